// HybridSamplerConv_65481071404168
// MI455X (gfx1250) — compile-verified
//
#include <hip/hip_runtime.h>

typedef __attribute__((ext_vector_type(2))) float v2f;
typedef __attribute__((ext_vector_type(4))) float v4f;
typedef __attribute__((ext_vector_type(8))) float v8f;

#define LOG2E 1.4426950408889634f

// sigmoid(x) = 1/(1+exp(-x)) via exp2 + rcp -> trans pipe, co-executes with VALU
__device__ __forceinline__ float fast_sigmoid(float x) {
    float e = __builtin_amdgcn_exp2f(-x * LOG2E);
    return __builtin_amdgcn_rcpf(1.0f + e);
}
// tanh(x) = 2*sigmoid(2x) - 1
__device__ __forceinline__ float fast_tanh(float x) {
    return __builtin_fmaf(2.0f, fast_sigmoid(2.0f * x), -1.0f);
}

__global__ void __launch_bounds__(256)
hybrid_sampler_conv_kernel(const float* __restrict__ inputs,   // [B,2]
                           const float* __restrict__ data,     // [B,2,2]
                           const float* __restrict__ conv_w,   // [2,2]
                           const float* __restrict__ conv_b,   // scalar
                           const float* __restrict__ w1,       // [3,4]
                           const float* __restrict__ b1,       // [4]
                           const float* __restrict__ w2,       // [4,2]
                           const float* __restrict__ b2,       // [2]
                           float* __restrict__ out,            // [B,2]
                           int n) {
    const int lane   = (int)(threadIdx.x & 31u);
    const int lane16 = lane & 15;
    const int half   = lane >> 4;

    // ---- wave-uniform weight loads (scalar path; tiny, L2-resident) ----
    const float cw0 = conv_w[0], cw1 = conv_w[1], cw2 = conv_w[2], cw3 = conv_w[3];
    const float cb  = conv_b[0];
    const float w200 = w2[0], w201 = w2[1];   // w2[i][j] = w2[i*2+j]
    const float w210 = w2[2], w211 = w2[3];
    const float w220 = w2[4], w221 = w2[5];
    const float w230 = w2[6], w231 = w2[7];
    const float b20 = b2[0], b21 = b2[1];

    // ---- A-matrix (loop invariant): A[m][k] = (k<3 ? w1[k][m] : b1[m]) for m<4, else 0
    // 32-bit 16x4 A layout: lane<16 holds (K0,K1), lane>=16 holds (K2,K3) of row lane%16
    v2f A = {0.0f, 0.0f};
    if (lane16 < 4) {
        if (half) { A.x = w1[2 * 4 + lane16]; A.y = b1[lane16]; }
        else      { A.x = w1[0 * 4 + lane16]; A.y = w1[1 * 4 + lane16]; }
    }

    const unsigned waveId = (blockIdx.x * blockDim.x + threadIdx.x) >> 5;

    #pragma unroll
    for (int it = 0; it < 4; ++it) {
        const int n0 = (int)(waveId * 64u) + it * 16;   // wave-uniform tile base
        if (n0 >= n) break;                              // uniform exit, EXEC stays full
        const int m = (n0 + lane16 < n) ? (n0 + lane16) : (n - 1);

        // prefetch next tile of the dominant stream (guarded, in-bounds only)
        if (n0 + 32 <= n) __builtin_prefetch(data + (size_t)(n0 + 16) * 4, 0, 1);

        // B VGPR0: lanes0-15 = x0[m], lanes16-31 = x1[m] (coalesced b32)
        const float bx = inputs[(size_t)m * 2 + half];

        // conv path: every lane loads its sample's 2x2 patch (b128) and computes
        // sigmoid(dot) -- high lanes' result is replaced by the bias-row 1.0
        const v4f d = *(const v4f*)(data + (size_t)m * 4);
        float logit = d.x * cw0;
        logit = __builtin_fmaf(d.y, cw1, logit);
        logit = __builtin_fmaf(d.z, cw2, logit);
        logit = __builtin_fmaf(d.w, cw3, logit);
        logit += cb;
        const float s  = fast_sigmoid(logit);
        const float by = half ? 1.0f : s;     // B VGPR1: K1=conv_out / K3=ones

        v2f Bv = {bx, by};
        v8f c = {};
        // D[m_row][n_col] = sum_k A[m_row][k]*B[k][n_col]  => h_pre[sample=n_col][m_row]
        c = __builtin_amdgcn_wmma_f32_16x16x4_f32(false, A, false, Bv,
                                                  (short)0, c, false, false);

        // lanes 0-15: c[0..3] hold this sample's 4 pre-activations
        const float h0 = fast_tanh(c[0]);
        const float h1 = fast_tanh(c[1]);
        const float h2 = fast_tanh(c[2]);
        const float h3 = fast_tanh(c[3]);

        float l0 = b20, l1 = b21;
        l0 = __builtin_fmaf(h0, w200, l0);  l1 = __builtin_fmaf(h0, w201, l1);
        l0 = __builtin_fmaf(h1, w210, l0);  l1 = __builtin_fmaf(h1, w211, l1);
        l0 = __builtin_fmaf(h2, w220, l0);  l1 = __builtin_fmaf(h2, w221, l1);
        l0 = __builtin_fmaf(h3, w230, l0);  l1 = __builtin_fmaf(h3, w231, l1);

        // 2-way softmax == sigmoid of the logit difference
        const float p0 = fast_sigmoid(l0 - l1);
        const float p1 = 1.0f - p0;

        if (half == 0 && (n0 + lane16) < n) {
            v2f r = {p0, p1};
            *(v2f*)(out + (size_t)(n0 + lane16) * 2) = r;   // coalesced b64 store
        }
    }
}

extern "C" void kernel_launch(void* const* d_in, const int* in_sizes, int n_in,
                              void* d_out, int out_size, void* d_ws, size_t ws_size,
                              hipStream_t stream) {
    const float* inputs = (const float*)d_in[0];
    const float* data   = (const float*)d_in[1];
    const float* conv_w = (const float*)d_in[2];
    const float* conv_b = (const float*)d_in[3];
    const float* w1     = (const float*)d_in[4];
    const float* b1     = (const float*)d_in[5];
    const float* w2     = (const float*)d_in[6];
    const float* b2     = (const float*)d_in[7];
    float* out = (float*)d_out;

    const int n = in_sizes[0] / 2;                 // B
    const int samplesPerBlock = 8 /*waves*/ * 64;  // 8 waves x (4 tiles x 16 samples)
    const int blocks = (n + samplesPerBlock - 1) / samplesPerBlock;

    hybrid_sampler_conv_kernel<<<blocks, 256, 0, stream>>>(
        inputs, data, conv_w, conv_b, w1, b1, w2, b2, out, n);
}